// RPN_3D_loss_17763984736675
// MI455X (gfx1250) — compile-verified
//
#include <hip/hip_runtime.h>
#include <math.h>

#define N_FG   8192
#define PATCH  32
#define PTS    (PATCH * PATCH)   // 1024
#define H_IM   512
#define W_IM   1760
#define PLANE  (H_IM * W_IM)
#define BOX_BLOCKS 32

typedef __attribute__((ext_vector_type(2))) float v2f;
typedef __attribute__((ext_vector_type(8))) float v8f;

// Bilinear sample with zeros padding of a planar [3,H,W] image at (ix,iy)
// (pixel coords, align_corners=False convention already folded in: ix = u-0.5).
__device__ __forceinline__ void sample3(const float* __restrict__ img,
                                        float ix, float iy, float r[3]) {
  float x0f = floorf(ix), y0f = floorf(iy);
  float wx1 = ix - x0f,  wy1 = iy - y0f;
  float wx0 = 1.0f - wx1, wy0 = 1.0f - wy1;
  int x0 = (int)x0f, y0 = (int)y0f;
  float wgt[2][2] = {{wx0 * wy0, wx1 * wy0}, {wx0 * wy1, wx1 * wy1}};
  r[0] = r[1] = r[2] = 0.0f;
#pragma unroll
  for (int cy = 0; cy < 2; ++cy) {
    int yy = y0 + cy;
    bool vy = (yy >= 0) && (yy <= H_IM - 1);
    int yc = min(max(yy, 0), H_IM - 1);
#pragma unroll
    for (int cx = 0; cx < 2; ++cx) {
      int xx = x0 + cx;
      bool vx = (xx >= 0) && (xx <= W_IM - 1);
      int xc = min(max(xx, 0), W_IM - 1);
      float w = wgt[cy][cx] * (float)(vx && vy);   // branchless zeros-pad
      int off = yc * W_IM + xc;
      r[0] += w * img[off];
      r[1] += w * img[off + PLANE];
      r[2] += w * img[off + 2 * PLANE];
    }
  }
}

// Photometric loss: one block per box; 8 waves; each wave handles 16 points
// per WMMA. Lanes 0-15 sample curr_image, lanes 16-31 sample prev_image of
// the SAME 16 points (C/D row replication trick), paired via shfl_xor(16).
__global__ void __launch_bounds__(256)
photo_kernel(const float* __restrict__ curr, const float* __restrict__ prev,
             const float* __restrict__ p2,   const float* __restrict__ p2inv,
             const float* __restrict__ mesh, const float* __restrict__ zproj,
             const float* __restrict__ motion, float* __restrict__ partial) {
  const int box  = blockIdx.x;
  const int tid  = threadIdx.x;
  const int lane = tid & 31;
  const int wave = tid >> 5;
  const int m    = lane & 15;        // matrix column / point-in-group
  const bool hi  = lane >= 16;       // hi half = prev-image role

  // Small camera matrices, per-thread registers.
  float P2[12], PI[12];
#pragma unroll
  for (int i = 0; i < 12; ++i) { P2[i] = p2[i]; PI[i] = p2inv[i]; }
  // M3 = P2 @ pinv(P2)  (3x3): proj = z*(M3 @ [u,v,1]) + P2[:, :3] @ motion
  float M3[9];
#pragma unroll
  for (int r = 0; r < 3; ++r)
#pragma unroll
    for (int c = 0; c < 3; ++c) {
      float s = 0.0f;
#pragma unroll
      for (int j = 0; j < 4; ++j) s += P2[r * 4 + j] * PI[j * 3 + c];
      M3[r * 3 + c] = s;
    }
  const float z  = zproj[box];
  const float m0 = motion[box * 3 + 0], m1 = motion[box * 3 + 1],
              m2 = motion[box * 3 + 2];
  const float tv[3] = {P2[0] * m0 + P2[1] * m1 + P2[2]  * m2,
                       P2[4] * m0 + P2[5] * m1 + P2[6]  * m2,
                       P2[8] * m0 + P2[9] * m1 + P2[10] * m2};

  // A operand (16x4 fp32): rows 0-2 AND 8-10 = [M3 | t]  -> D rows replicate
  // the projection into both lane halves. Lane layout per ISA: lane<16 holds
  // cols (0,1), lane>=16 holds cols (2,3) of row m.
  const int eff = (m < 3) ? m : ((m >= 8 && m < 11) ? (m - 8) : -1);
  v2f a; a.x = 0.0f; a.y = 0.0f;
  if (eff >= 0) {
    a.x = hi ? M3[eff * 3 + 2] : M3[eff * 3 + 0];
    a.y = hi ? tv[eff]         : M3[eff * 3 + 1];
  }

  const float* __restrict__ img = hi ? prev : curr;
  float acc = 0.0f;

  for (int it = 0; it < 8; ++it) {
    const int grp = wave * 8 + it;            // 0..63 group of 16 points
    const int p   = grp * 16 + m;             // point index within patch
    const int base = (box * PTS + p) * 3;
    const float u = mesh[base], v = mesh[base + 1];   // third comp == 1
    if (it < 7)
      __builtin_prefetch((const void*)(mesh + base + 48), 0, 1); // next group

    // B operand (4x16 fp32): column p_m = [u*z, v*z, z, 1]^T.
    v2f b;
    b.x = hi ? z    : u * z;
    b.y = hi ? 1.0f : v * z;
    v8f c = {};
    v8f d = __builtin_amdgcn_wmma_f32_16x16x4_f32(
        /*neg_a=*/false, a, /*neg_b=*/false, b,
        /*c_mod=*/(short)0, c, /*reuse_a=*/false, /*reuse_b=*/false);
    const float prx = d[0], pry = d[1], prz = d[2];   // proj of point m

    // Sampling coords: curr lanes use raw (u,v); prev lanes use projection.
    const float invz = 1.0f / prz;
    const float sx = hi ? prx * invz : u;
    const float sy = hi ? pry * invz : v;
    float s[3];
    sample3(img, sx - 0.5f, sy - 0.5f, s);

    float dsum = 0.0f;
#pragma unroll
    for (int ch = 0; ch < 3; ++ch) {
      const float o = __shfl_xor(s[ch], 16, 32);      // pair curr <-> prev
      dsum += fabsf(s[ch] - o);
    }
    acc += hi ? dsum : 0.0f;                          // count each point once
  }

  // Deterministic block reduction: wave32 shfl tree + LDS.
#pragma unroll
  for (int off = 16; off > 0; off >>= 1) acc += __shfl_xor(acc, off, 32);
  __shared__ float sm[8];
  if (lane == 0) sm[wave] = acc;
  __syncthreads();
  if (tid == 0) {
    float s = 0.0f;
#pragma unroll
    for (int w = 0; w < 8; ++w) s += sm[w];
    partial[box] = s;
  }
}

// Box loss: one thread per box; 32 blocks of 256.
__global__ void __launch_bounds__(256)
box_kernel(const float* __restrict__ p2, const float* __restrict__ bbox3d,
           const float* __restrict__ tar, float* __restrict__ partial) {
  const int i = blockIdx.x * 256 + threadIdx.x;
  float P2[12];
#pragma unroll
  for (int k = 0; k < 12; ++k) P2[k] = p2[k];
  const float* bb = bbox3d + i * 7;
  const float ry = bb[0], l = bb[1], w = bb[2], h = bb[3],
              x  = bb[4], y = bb[5], z = bb[6];
  const float cs = cosf(ry), sn = sinf(ry);
  const float SX[8] = {-1, 1, 1, 1, 1, -1, -1, -1};
  const float SY[8] = {-1, -1, 1, 1, -1, -1, 1, 1};
  const float SZ[8] = {-1, -1, -1, 1, 1, -1, 1, -1};
  float mnx = 1e30f, mny = 1e30f, mxx = -1e30f, mxy = -1e30f;
#pragma unroll
  for (int k = 0; k < 8; ++k) {
    const float xc = SX[k] * l * 0.5f, yc = SY[k] * h * 0.5f,
                zc = SZ[k] * w * 0.5f;
    const float cx = cs * xc + sn * zc + x;
    const float cy = yc + y;
    const float cz = -sn * xc + cs * zc + z;
    const float px = P2[0] * cx + P2[1] * cy + P2[2]  * cz + P2[3];
    const float py = P2[4] * cx + P2[5] * cy + P2[6]  * cz + P2[7];
    const float pz = P2[8] * cx + P2[9] * cy + P2[10] * cz + P2[11];
    const float ux = px / pz, uy = py / pz;
    mnx = fminf(mnx, ux); mxx = fmaxf(mxx, ux);
    mny = fminf(mny, uy); mxy = fmaxf(mxy, uy);
  }
  const float* tg = tar + i * 4;
  const float bx[4] = {mnx, mny, mxx, mxy};
  float s = 0.0f;
#pragma unroll
  for (int k = 0; k < 4; ++k) {
    const float dd = fabsf(bx[k] - tg[k]);
    s += (dd < 1.0f) ? 0.5f * dd * dd : dd - 0.5f;   // smooth L1
  }
  const int lane = threadIdx.x & 31, wave = threadIdx.x >> 5;
#pragma unroll
  for (int off = 16; off > 0; off >>= 1) s += __shfl_xor(s, off, 32);
  __shared__ float sm[8];
  if (lane == 0) sm[wave] = s;
  __syncthreads();
  if (threadIdx.x == 0) {
    float t = 0.0f;
#pragma unroll
    for (int w8 = 0; w8 < 8; ++w8) t += sm[w8];
    partial[N_FG + blockIdx.x] = t;
  }
}

// Final deterministic reduction + combine.
__global__ void __launch_bounds__(256)
finalize_kernel(const float* __restrict__ partial, float* __restrict__ out) {
  const int tid = threadIdx.x, lane = tid & 31, wave = tid >> 5;
  float s = 0.0f;
  for (int i = tid; i < N_FG; i += 256) s += partial[i];
#pragma unroll
  for (int off = 16; off > 0; off >>= 1) s += __shfl_xor(s, off, 32);
  __shared__ float sm[8];
  if (lane == 0) sm[wave] = s;
  __syncthreads();
  if (tid == 0) {
    float photo = 0.0f;
#pragma unroll
    for (int w = 0; w < 8; ++w) photo += sm[w];
    float boxs = 0.0f;
#pragma unroll
    for (int w = 0; w < BOX_BLOCKS; ++w) boxs += partial[N_FG + w];
    out[0] = photo / (3.0f * (float)N_FG * (float)PTS)
           + 0.1f * (boxs / ((float)N_FG * 4.0f));
  }
}

extern "C" void kernel_launch(void* const* d_in, const int* in_sizes, int n_in,
                              void* d_out, int out_size, void* d_ws, size_t ws_size,
                              hipStream_t stream) {
  (void)in_sizes; (void)n_in; (void)out_size; (void)ws_size;
  const float* curr   = (const float*)d_in[0];
  const float* prev   = (const float*)d_in[1];
  const float* p2     = (const float*)d_in[2];
  const float* p2inv  = (const float*)d_in[3];
  const float* mesh   = (const float*)d_in[4];
  const float* zproj  = (const float*)d_in[5];
  const float* motion = (const float*)d_in[6];
  const float* bbox3d = (const float*)d_in[7];
  const float* tar    = (const float*)d_in[8];
  float* out = (float*)d_out;
  float* partial = (float*)d_ws;     // [0,N_FG): photo partials; [N_FG,N_FG+32): box

  photo_kernel<<<N_FG, 256, 0, stream>>>(curr, prev, p2, p2inv, mesh, zproj,
                                         motion, partial);
  box_kernel<<<BOX_BLOCKS, 256, 0, stream>>>(p2, bbox3d, tar, partial);
  finalize_kernel<<<1, 256, 0, stream>>>(partial, out);
}